// SelfAttention_59416577573035
// MI455X (gfx1250) — compile-verified
//
#include <hip/hip_runtime.h>

// Fused kernel-gated neighbor attention, fp32, CDNA5 (gfx1250, wave32).
// Two kernels:
//   1) query_proj: q = target @ Wq^T + bq   ([B,128]x[128,128] WMMA GEMM,
//      TDM-staged 16-row tiles, double-buffered; result stored into d_out,
//      which is exactly B*H floats and is re-read then overwritten by (2)).
//   2) fused_attn: per node b --
//      keys   = neighs @ Wk^T + bk          (V_WMMA_F32_16X16X4_F32)
//      values = neighs @ Wv^T + bv          (V_WMMA_F32_16X16X4_F32)
//      scores = (q.k)/sqrt(H)*kernels ; attn = softmax ; out = attn @ values
// Memory-bound: ~900 MB compulsory -> ~39 us floor @ 23.3 TB/s. Per-node LDS
// tiles are double-buffered and staged by the Tensor Data Mover (TENSORcnt),
// overlapping node i+1's DMA with node i's WMMA. Weights live in registers.

#define HDIM   128
#define KN     32
#define BNODES 50000
#define GRID   2048
#define QTILES (BNODES / 16)   // 3125 exact 16-row tiles
#define QGRID  1024
#define A_PITCH 132   // TDM pad: every 128 DWORDs insert 4 DWORDs -> pitch 132

typedef __attribute__((ext_vector_type(2))) float    v2f;
typedef __attribute__((ext_vector_type(8))) float    v8f;
typedef __attribute__((ext_vector_type(4))) unsigned v4u;
typedef __attribute__((ext_vector_type(8))) int      v8i;
typedef __attribute__((ext_vector_type(4))) int      v4i;

__device__ __forceinline__ unsigned lds_off(const void* p) {
    // generic LDS pointer: aperture lives in addr[63:32], addr[31:0] is the
    // workgroup-relative LDS byte offset (ISA 10.2 aperture rules).
    return (unsigned)(unsigned long long)p;
}

// 2D TDM load: tile (tile_d1 rows x tile_d0 elems, f32) from a row-major
// tensor (stride0 elems/row) into LDS, optional 4-DWORD pad per 128 DWORDs.
__device__ __forceinline__ void tdm_load_2d(unsigned lds, const void* gptr,
                                            unsigned tensor_d0, unsigned tensor_d1,
                                            unsigned tile_d0, unsigned tile_d1,
                                            unsigned stride0, int pad)
{
    const unsigned long long ga = (unsigned long long)gptr;
    v4u g0;
    g0[0] = 1u;                                        // count=1 (valid), user mode
    g0[1] = lds;                                       // lds_addr [63:32]
    g0[2] = (unsigned)ga;                              // global_addr[31:0]
    g0[3] = (unsigned)((ga >> 32) & 0x01FFFFFFull)     // global_addr[56:32]
          | (2u << 30);                                // type=2 ("image")
    v8i g1;
    unsigned w0 = (2u << 16);                          // data_size=2 -> 4 bytes
    if (pad) w0 |= (1u << 20)                          // pad_enable
               |  (6u << 22)                           // pad_interval: 128 DWORDs
               |  (3u << 25);                          // pad_amount : 4 DWORDs
    g1[0] = (int)w0;
    g1[1] = (int)((tensor_d0 & 0xFFFFu) << 16);        // tensor_dim0[15:0] @ [63:48]
    g1[2] = (int)(((tensor_d1 & 0xFFFFu) << 16) | (tensor_d0 >> 16));
    g1[3] = (int)(((tile_d0 & 0xFFFFu) << 16) | (tensor_d1 >> 16)); // tile_dim0 @127:112
    g1[4] = (int)(tile_d1 & 0xFFFFu);                  // tile_dim1, tile_dim2=0
    g1[5] = (int)stride0;                              // dim0_stride[31:0]
    g1[6] = 0;
    g1[7] = 0;
    const v4i z4 = {0, 0, 0, 0};
    const v8i z8 = {0, 0, 0, 0, 0, 0, 0, 0};
    // amdgpu-toolchain (clang-23) 6-arg form: (g0, g1, g2, g3, g4, cpol)
    __builtin_amdgcn_tensor_load_to_lds(g0, g1, z4, z4, z8, 0);
}

// ---------------------------------------------------------------------------
// Kernel 1: q = target @ Wq^T + bq, written to qs (aliases d_out).
// ---------------------------------------------------------------------------
__launch_bounds__(256)
__global__ void query_proj_kernel(const float* __restrict__ target,
                                  const float* __restrict__ Wq,
                                  const float* __restrict__ bq,
                                  float* qs)
{
    __shared__ __align__(16) float sTile[2][16 * A_PITCH];

    const int tid  = threadIdx.x;
    const int wave = tid >> 5;
    const int lane = tid & 31;
    const int c    = lane & 15;
    const int hi   = lane >> 4;
    const int col  = (wave << 4) + c;

    v2f wqf[32];
    {
        const float* wqrow = Wq + col * HDIM;
#pragma unroll
        for (int kk = 0; kk < 32; ++kk)
            wqf[kk] = *reinterpret_cast<const v2f*>(wqrow + 4 * kk + 2 * hi);
    }
    const float bq_l = bq[col];

    auto prefetch = [&](int t, int buf) {
        tdm_load_2d(lds_off(&sTile[buf][0]), target + (size_t)t * 16 * HDIM,
                    HDIM, 16, HDIM, 16, HDIM, /*pad=*/1);
    };

    int t = blockIdx.x;                       // QGRID <= QTILES
    if (wave == 0) prefetch(t, 0);
    int cur = 0;
    for (; t < QTILES; t += QGRID, cur ^= 1) {
        const int nt = t + QGRID;
        if (wave == 0) {
            if (nt < QTILES) {
                prefetch(nt, cur ^ 1);
                __builtin_amdgcn_s_wait_tensorcnt((short)1);
            } else {
                __builtin_amdgcn_s_wait_tensorcnt((short)0);
            }
        }
        __syncthreads();

        v8f acc = {bq_l, bq_l, bq_l, bq_l, bq_l, bq_l, bq_l, bq_l};
        const float* aBase = &sTile[cur][c * A_PITCH];
#pragma unroll
        for (int kk = 0; kk < 32; ++kk) {
            const v2f a = *reinterpret_cast<const v2f*>(aBase + 4 * kk + 2 * hi);
            acc = __builtin_amdgcn_wmma_f32_16x16x4_f32(false, a, false, wqf[kk],
                                                        (short)0, acc, false, false);
        }
        float* orow = qs + ((size_t)t * 16 + 8 * hi) * HDIM + col;
#pragma unroll
        for (int j = 0; j < 8; ++j)           // acc elem j -> row j + 8*hi
            orow[(size_t)j * HDIM] = acc[j];
        __syncthreads();                      // all reads done before buffer reuse
    }
}

// ---------------------------------------------------------------------------
// Kernel 2: fused attention. qs aliases out (row b is read, then overwritten).
// ---------------------------------------------------------------------------
__launch_bounds__(256)
__global__ void fused_attn_kernel(const float* __restrict__ neighs,
                                  const float* __restrict__ kernels,
                                  const float* __restrict__ Wk,
                                  const float* __restrict__ bk,
                                  const float* __restrict__ Wv,
                                  const float* __restrict__ bv,
                                  const float* qs,
                                  float* out)
{
    __shared__ __align__(16) float sA[2][KN * A_PITCH];  // neighs tiles (padded)
    __shared__ __align__(16) float sQ[2][HDIM];          // q rows (from kernel 1)
    __shared__ __align__(16) float sKer[2][KN];          // kernel gates
    __shared__ __align__(16) float sScores[KN];
    __shared__ __align__(16) float sAttn[KN];

    const int tid  = threadIdx.x;
    const int wave = tid >> 5;         // 0..7 : owns output cols [16w, 16w+16)
    const int lane = tid & 31;
    const int c    = lane & 15;
    const int hi   = lane >> 4;
    const int col  = (wave << 4) + c;

    // ---- Register-resident B-fragments of Wk^T / Wv^T (once per block) ----
    v2f wkf[32], wvf[32];
    {
        const float* wkrow = Wk + col * HDIM;
        const float* wvrow = Wv + col * HDIM;
#pragma unroll
        for (int kk = 0; kk < 32; ++kk) {
            const int off = 4 * kk + 2 * hi;
            wkf[kk] = *reinterpret_cast<const v2f*>(wkrow + off);
            wvf[kk] = *reinterpret_cast<const v2f*>(wvrow + off);
        }
    }
    const float bk_l = bk[col];
    const float bv_l = bv[col];
    const float inv_sqrt_h = 0.08838834764831845f; // 1/sqrt(128)

    if (tid < KN) sScores[tid] = 0.f;

    // Wave 0 issues 3 TDM descriptors per node: neighs tile, q row, kernels.
    auto tdm_prefetch = [&](int b, int buf) {
        tdm_load_2d(lds_off(&sA[buf][0]), neighs + (size_t)b * (KN * HDIM),
                    HDIM, KN, HDIM, KN, HDIM, /*pad=*/1);
        tdm_load_2d(lds_off(&sQ[buf][0]), qs + (size_t)b * HDIM,
                    HDIM, 1, HDIM, 1, HDIM, /*pad=*/0);
        tdm_load_2d(lds_off(&sKer[buf][0]), kernels + (size_t)b * KN,
                    KN, 1, KN, 1, KN, /*pad=*/0);
    };

    int b = blockIdx.x;                 // GRID <= BNODES: every block has work
    if (wave == 0) tdm_prefetch(b, 0);
    int cur = 0;
    for (; b < BNODES; b += GRID, cur ^= 1) {
        const int nb = b + GRID;
        // ---- pipeline: issue node i+1, wait for node i (TDM is in-order) ----
        if (wave == 0) {
            if (nb < BNODES) {
                tdm_prefetch(nb, cur ^ 1);
                __builtin_amdgcn_s_wait_tensorcnt((short)3);
            } else {
                __builtin_amdgcn_s_wait_tensorcnt((short)0);
            }
        }
        __syncthreads();                                    // A: tile visible

        // ---- phase 1: keys & values via V_WMMA_F32_16X16X4_F32 ----
        v8f ka0 = {bk_l, bk_l, bk_l, bk_l, bk_l, bk_l, bk_l, bk_l};
        v8f ka1 = ka0;
        v8f va0 = {bv_l, bv_l, bv_l, bv_l, bv_l, bv_l, bv_l, bv_l};
        v8f va1 = va0;
        const float* aBase0 = &sA[cur][c * A_PITCH];
        const float* aBase1 = &sA[cur][(16 + c) * A_PITCH];
#pragma unroll
        for (int kk = 0; kk < 32; ++kk) {
            const int off = 4 * kk + 2 * hi;
            const v2f a0 = *reinterpret_cast<const v2f*>(aBase0 + off);
            const v2f a1 = *reinterpret_cast<const v2f*>(aBase1 + off);
            ka0 = __builtin_amdgcn_wmma_f32_16x16x4_f32(false, a0, false, wkf[kk],
                                                        (short)0, ka0, false, false);
            ka1 = __builtin_amdgcn_wmma_f32_16x16x4_f32(false, a1, false, wkf[kk],
                                                        (short)0, ka1, false, false);
            va0 = __builtin_amdgcn_wmma_f32_16x16x4_f32(false, a0, false, wvf[kk],
                                                        (short)0, va0, false, false);
            va1 = __builtin_amdgcn_wmma_f32_16x16x4_f32(false, a1, false, wvf[kk],
                                                        (short)0, va1, false, false);
        }

        // ---- phase 2a: score partials from key accumulators ----
        // acc elem j in lane -> key row (j+8*hi [+16 for ka1]), col = col.
        // sScores was re-armed to zero before barrier D of the previous node.
        {
            const float qcol = sQ[cur][col];                // bias already folded
#pragma unroll
            for (int j = 0; j < 8; ++j) {
                float p0 = ka0[j] * qcol;
                float p1 = ka1[j] * qcol;
#pragma unroll
                for (int m = 8; m >= 1; m >>= 1) {          // butterfly over 16 cols
                    p0 += __shfl_xor(p0, m, 32);
                    p1 += __shfl_xor(p1, m, 32);
                }
                if (c == 0) {
                    atomicAdd(&sScores[j + 8 * hi], p0);
                    atomicAdd(&sScores[16 + j + 8 * hi], p1);
                }
            }
        }
        __syncthreads();                                    // C: scores complete
                                                            // (also fences sA reads
                                                            //  before buffer reuse)
        // ---- phase 2b: softmax on wave 0 (wave32 shfl reductions) ----
        if (tid < KN) {
            float s = sScores[tid] * inv_sqrt_h * sKer[cur][tid];
            sScores[tid] = 0.f;                             // re-arm for next node
            float mx = s;
            for (int off = 16; off >= 1; off >>= 1)
                mx = fmaxf(mx, __shfl_xor(mx, off, 32));
            const float e = __expf(s - mx);
            float sum = e;
            for (int off = 16; off >= 1; off >>= 1)
                sum += __shfl_xor(sum, off, 32);
            sAttn[tid] = e / sum;
        }
        __syncthreads();                                    // D: attn visible

        // ---- phase 3: out = attn @ values, straight from accumulators ----
        {
            const float4 a0 = *reinterpret_cast<const float4*>(&sAttn[8 * hi]);
            const float4 a1 = *reinterpret_cast<const float4*>(&sAttn[8 * hi + 4]);
            const float4 a2 = *reinterpret_cast<const float4*>(&sAttn[16 + 8 * hi]);
            const float4 a3 = *reinterpret_cast<const float4*>(&sAttn[16 + 8 * hi + 4]);
            float v = a0.x * va0[0] + a0.y * va0[1] + a0.z * va0[2] + a0.w * va0[3]
                    + a1.x * va0[4] + a1.y * va0[5] + a1.z * va0[6] + a1.w * va0[7]
                    + a2.x * va1[0] + a2.y * va1[1] + a2.z * va1[2] + a2.w * va1[3]
                    + a3.x * va1[4] + a3.y * va1[5] + a3.z * va1[6] + a3.w * va1[7];
            v += __shfl_xor(v, 16, 32);                     // combine col halves
            if (hi == 0) out[(size_t)b * HDIM + col] = v;   // overwrites q row b
        }
        // Next iteration's TDM writes into sA/sQ/sKer[cur^1] are safe: all reads
        // of that buffer completed before barrier D of the previous iteration.
    }
}

extern "C" void kernel_launch(void* const* d_in, const int* in_sizes, int n_in,
                              void* d_out, int out_size, void* d_ws, size_t ws_size,
                              hipStream_t stream) {
    (void)in_sizes; (void)n_in; (void)d_ws; (void)ws_size; (void)out_size;
    const float* target  = (const float*)d_in[0];
    const float* neighs  = (const float*)d_in[1];
    const float* kernels = (const float*)d_in[2];
    const float* Wq      = (const float*)d_in[3];
    const float* bq      = (const float*)d_in[4];
    const float* Wk      = (const float*)d_in[5];
    const float* bk      = (const float*)d_in[6];
    const float* Wv      = (const float*)d_in[7];
    const float* bv      = (const float*)d_in[8];
    float* out = (float*)d_out;

    // Stage 1: q -> d_out (B*H floats). Stage 2 reads row b then overwrites it.
    query_proj_kernel<<<QGRID, 256, 0, stream>>>(target, Wq, bq, out);
    fused_attn_kernel<<<GRID, 256, 0, stream>>>(neighs, kernels,
                                                Wk, bk, Wv, bv, out, out);
}